// Model_6571299963067
// MI455X (gfx1250) — compile-verified
//
#include <hip/hip_runtime.h>
#include <hip/hip_bf16.h>

typedef __attribute__((ext_vector_type(2))) float v2f;
typedef __attribute__((ext_vector_type(8))) float v8f;

#define HDIM 64

// ---------------------------------------------------------------------------
// Degree counts (edge structure is shared by both layers -> compute once).
// ---------------------------------------------------------------------------
__global__ __launch_bounds__(256)
void deg_kernel(const int* __restrict__ src, const int* __restrict__ dst,
                float* __restrict__ degU, float* __restrict__ degM, int nE) {
    int e = blockIdx.x * blockDim.x + threadIdx.x;
    if (e >= nE) return;
    atomicAdd(degU + src[e], 1.0f);
    atomicAdd(degM + dst[e], 1.0f);
}

// ---------------------------------------------------------------------------
// x_movie = movie_x @ lin_W.T + lin_b + movie_emb   ([M,20]x[20,64], K=20)
// ---------------------------------------------------------------------------
__global__ __launch_bounds__(256)
void movie_init_kernel(const float* __restrict__ mx, const float* __restrict__ linW,
                       const float* __restrict__ linb, const float* __restrict__ memb,
                       float* __restrict__ XM, int M) {
    int t = blockIdx.x * blockDim.x + threadIdx.x;
    if (t >= M * HDIM) return;
    int m = t >> 6, h = t & 63;
    float acc = linb[h] + memb[t];
    const float* xr = mx + (size_t)m * 20;
    const float* wr = linW + (size_t)h * 20;
#pragma unroll
    for (int f = 0; f < 20; ++f) acc = fmaf(xr[f], wr[f], acc);
    XM[t] = acc;
}

// ---------------------------------------------------------------------------
// Segment-sum scatter: agg[dst[e], :] += x_src[src[e], :]
// 16 threads per edge, float4 per thread, f32 global atomics.
// ---------------------------------------------------------------------------
__global__ __launch_bounds__(256)
void scatter_kernel(const float* __restrict__ xsrc, const int* __restrict__ src,
                    const int* __restrict__ dst, float* __restrict__ agg, int nE) {
    long long tid = (long long)blockIdx.x * blockDim.x + threadIdx.x;
    if (tid >= (long long)nE * 16) return;
    int e = (int)(tid >> 4);
    int q = (int)(tid & 15);
    int s = src[e], d = dst[e];
    const float4 v = *(const float4*)(xsrc + (size_t)s * HDIM + q * 4);
    float* p = agg + (size_t)d * HDIM + q * 4;
    atomicAdd(p + 0, v.x);
    atomicAdd(p + 1, v.y);
    atomicAdd(p + 2, v.z);
    atomicAdd(p + 3, v.w);
}

// ---------------------------------------------------------------------------
// SAGE combine: out = (agg/max(deg,1)) @ Wl.T + xdst @ Wr.T + bias [, ReLU]
// One block = 16 rows x 64 cols; 4 waves, each owns a 16x16 tile.
// Fused K=128 chain of V_WMMA_F32_16X16X4_F32 (32 wmma per tile).
// n is a multiple of 16 -> no tails; EXEC is all-ones throughout (WMMA req).
// ---------------------------------------------------------------------------
__global__ __launch_bounds__(128)
void combine_wmma_kernel(const float* __restrict__ agg, const float* __restrict__ deg,
                         const float* __restrict__ xdst,
                         const float* __restrict__ Wl, const float* __restrict__ Wr,
                         const float* __restrict__ bias, float* __restrict__ out,
                         int do_relu) {
    __shared__ float A1[16][68];   // mean tile, pitch 68 to avoid bank conflicts
    __shared__ float A2[16][68];   // x_dst tile

    const int row0 = blockIdx.x * 16;
    const int tid  = threadIdx.x;          // 0..127

    // Cooperative stage: 128 threads x 8 floats = 16x64 tile per matrix.
    {
        int r  = tid >> 3;                 // 0..15
        int c0 = (tid & 7) * 8;            // 0,8,..,56
        float dinv = 1.0f / fmaxf(deg[row0 + r], 1.0f);
        const float* ga = agg  + (size_t)(row0 + r) * HDIM + c0;
        const float* gx = xdst + (size_t)(row0 + r) * HDIM + c0;
#pragma unroll
        for (int j = 0; j < 8; ++j) {
            A1[r][c0 + j] = ga[j] * dinv;
            A2[r][c0 + j] = gx[j];
        }
    }
    __syncthreads();

    const int wave = tid >> 5;             // 0..3  -> column tile
    const int lane = tid & 31;
    const int m    = lane & 15;            // A row / B col / D col
    const int half = lane >> 4;            // 0 or 1
    const int col0 = wave * 16;

    v8f c = {};
#pragma unroll
    for (int kk = 0; kk < 16; ++kk) {      // mean x Wl.T
        int k = kk * 4 + half * 2;
        v2f a, b;
        a.x = A1[m][k];  a.y = A1[m][k + 1];
        const float* wp = Wl + (size_t)(col0 + m) * HDIM + k;
        b.x = wp[0];     b.y = wp[1];
        c = __builtin_amdgcn_wmma_f32_16x16x4_f32(false, a, false, b,
                                                  (short)0, c, false, false);
    }
#pragma unroll
    for (int kk = 0; kk < 16; ++kk) {      // x_dst x Wr.T (same accumulator)
        int k = kk * 4 + half * 2;
        v2f a, b;
        a.x = A2[m][k];  a.y = A2[m][k + 1];
        const float* wp = Wr + (size_t)(col0 + m) * HDIM + k;
        b.x = wp[0];     b.y = wp[1];
        c = __builtin_amdgcn_wmma_f32_16x16x4_f32(false, a, false, b,
                                                  (short)0, c, false, false);
    }

    // D layout: VGPR r -> row (row0 + r + 8*half), col (col0 + m)
    float bv = bias[col0 + m];
#pragma unroll
    for (int r = 0; r < 8; ++r) {
        float v = c[r] + bv;
        if (do_relu) v = fmaxf(v, 0.0f);
        out[(size_t)(row0 + r + 8 * half) * HDIM + col0 + m] = v;
    }
}

// ---------------------------------------------------------------------------
// out[l] = dot(U2[eli_user[l]], M2[eli_movie[l]]), H=64.
// 8 lanes per pair (2x float4 each side), xor-shuffle reduce.
// ---------------------------------------------------------------------------
__global__ __launch_bounds__(256)
void edge_dot_kernel(const float* __restrict__ U2, const float* __restrict__ M2,
                     const int* __restrict__ eu, const int* __restrict__ em,
                     float* __restrict__ out, int L) {
    long long tid = (long long)blockIdx.x * blockDim.x + threadIdx.x;
    if (tid >= (long long)L * 8) return;
    int p = (int)(tid >> 3);
    int s = (int)(tid & 7);
    const float4* a = (const float4*)(U2 + (size_t)eu[p] * HDIM + s * 8);
    const float4* b = (const float4*)(M2 + (size_t)em[p] * HDIM + s * 8);
    float4 a0 = a[0], a1 = a[1], b0 = b[0], b1 = b[1];
    float acc = a0.x * b0.x + a0.y * b0.y + a0.z * b0.z + a0.w * b0.w
              + a1.x * b1.x + a1.y * b1.y + a1.z * b1.z + a1.w * b1.w;
    acc += __shfl_xor(acc, 1);
    acc += __shfl_xor(acc, 2);
    acc += __shfl_xor(acc, 4);
    if (s == 0) out[p] = acc;
}

// ---------------------------------------------------------------------------
extern "C" void kernel_launch(void* const* d_in, const int* in_sizes, int n_in,
                              void* d_out, int out_size, void* d_ws, size_t ws_size,
                              hipStream_t stream) {
    (void)n_in; (void)ws_size;
    const float* movie_x   = (const float*)d_in[2];
    const int*   e_src     = (const int*)d_in[3];
    const int*   e_dst     = (const int*)d_in[4];
    const int*   eli_u     = (const int*)d_in[5];
    const int*   eli_m     = (const int*)d_in[6];
    const float* user_emb  = (const float*)d_in[7];
    const float* movie_emb = (const float*)d_in[8];
    const float* lin_W     = (const float*)d_in[9];
    const float* lin_b     = (const float*)d_in[10];
    const float* W1um_l    = (const float*)d_in[11];
    const float* W1um_r    = (const float*)d_in[12];
    const float* W1mu_l    = (const float*)d_in[13];
    const float* W1mu_r    = (const float*)d_in[14];
    const float* W2um_l    = (const float*)d_in[15];
    const float* W2um_r    = (const float*)d_in[16];
    const float* W2mu_l    = (const float*)d_in[17];
    const float* W2mu_r    = (const float*)d_in[18];
    const float* b1um      = (const float*)d_in[19];
    const float* b1mu      = (const float*)d_in[20];
    const float* b2um      = (const float*)d_in[21];
    const float* b2mu      = (const float*)d_in[22];
    float* out = (float*)d_out;

    const int U = in_sizes[7] / HDIM;   // 100000 (multiple of 16)
    const int M = in_sizes[8] / HDIM;   // 50000  (multiple of 16)
    const int E = in_sizes[3];          // 2500000
    const int L = in_sizes[5];          // 1000000
    (void)out_size;

    // Workspace carve-out (256B aligned).
    size_t off = 0;
    auto carve = [&](size_t nfloats) {
        float* p = (float*)((char*)d_ws + off);
        off += ((nfloats * sizeof(float) + 255) / 256) * 256;
        return p;
    };
    float* XM    = carve((size_t)M * HDIM);   // initial movie features
    float* AGG_M = carve((size_t)M * HDIM);
    float* AGG_U = carve((size_t)U * HDIM);
    float* H_M   = carve((size_t)M * HDIM);   // layer-1 movie output (relu)
    float* H_U   = carve((size_t)U * HDIM);   // layer-1 user output (relu)
    float* M2    = carve((size_t)M * HDIM);   // layer-2 movie output
    float* U2    = carve((size_t)U * HDIM);   // layer-2 user output
    float* DEG_M = carve((size_t)M);
    float* DEG_U = carve((size_t)U);

    const int TB = 256;
    dim3 b(TB);
    long long sc_threads = (long long)E * 16;
    int sc_blocks = (int)((sc_threads + TB - 1) / TB);

    // Degrees (once).
    hipMemsetAsync(DEG_M, 0, (size_t)M * sizeof(float), stream);
    hipMemsetAsync(DEG_U, 0, (size_t)U * sizeof(float), stream);
    deg_kernel<<<(E + TB - 1) / TB, b, 0, stream>>>(e_src, e_dst, DEG_U, DEG_M, E);

    // Movie feature init.
    movie_init_kernel<<<(M * HDIM + TB - 1) / TB, b, 0, stream>>>(
        movie_x, lin_W, lin_b, movie_emb, XM, M);

    // ---------------- Layer 1 ----------------
    hipMemsetAsync(AGG_M, 0, (size_t)M * HDIM * sizeof(float), stream);
    hipMemsetAsync(AGG_U, 0, (size_t)U * HDIM * sizeof(float), stream);
    scatter_kernel<<<sc_blocks, b, 0, stream>>>(user_emb, e_src, e_dst, AGG_M, E);
    scatter_kernel<<<sc_blocks, b, 0, stream>>>(XM, e_dst, e_src, AGG_U, E);
    combine_wmma_kernel<<<M / 16, 128, 0, stream>>>(AGG_M, DEG_M, XM,
                                                    W1um_l, W1um_r, b1um, H_M, 1);
    combine_wmma_kernel<<<U / 16, 128, 0, stream>>>(AGG_U, DEG_U, user_emb,
                                                    W1mu_l, W1mu_r, b1mu, H_U, 1);

    // ---------------- Layer 2 ----------------
    hipMemsetAsync(AGG_M, 0, (size_t)M * HDIM * sizeof(float), stream);
    hipMemsetAsync(AGG_U, 0, (size_t)U * HDIM * sizeof(float), stream);
    scatter_kernel<<<sc_blocks, b, 0, stream>>>(H_U, e_src, e_dst, AGG_M, E);
    scatter_kernel<<<sc_blocks, b, 0, stream>>>(H_M, e_dst, e_src, AGG_U, E);
    combine_wmma_kernel<<<M / 16, 128, 0, stream>>>(AGG_M, DEG_M, H_M,
                                                    W2um_l, W2um_r, b2um, M2, 0);
    combine_wmma_kernel<<<U / 16, 128, 0, stream>>>(AGG_U, DEG_U, H_U,
                                                    W2mu_l, W2mu_r, b2mu, U2, 0);

    // ---------------- Edge classifier ----------------
    long long dt = (long long)L * 8;
    edge_dot_kernel<<<(int)((dt + TB - 1) / TB), b, 0, stream>>>(U2, M2, eli_u, eli_m, out, L);
}